// TensorNetRepresentation_45930380264225
// MI455X (gfx1250) — compile-verified
//
#include <hip/hip_runtime.h>
#include <math.h>

#define N_ATOMS 25000
#define N_PAIRS 400000
// acc layout per atom: [comp 0..9][32 feats]; comp 0=I, 1..3=A(x,y,z), 4..9=S(xx,xy,xz,yy,yz,zz)
#define ACC_FLOATS (N_ATOMS * 320)

typedef float v2f __attribute__((ext_vector_type(2)));
typedef float v8f __attribute__((ext_vector_type(8)));

__device__ __forceinline__ v8f wmma4(v2f a, v2f b, v8f c) {
  // D(16x16,f32) = A(16x4,f32) x B(4x16,f32) + C   — full f32 precision
  return __builtin_amdgcn_wmma_f32_16x16x4_f32(false, a, false, b, (short)0, c, false, false);
}

__device__ __forceinline__ void atomAddF(float* p, float v) {
  __hip_atomic_fetch_add(p, v, __ATOMIC_RELAXED, __HIP_MEMORY_SCOPE_AGENT);
}

__global__ void zero_ws_kernel(float* __restrict__ p, int n) {
  int i = blockIdx.x * blockDim.x + threadIdx.x;
  if (i < n) p[i] = 0.0f;
}

__global__ __launch_bounds__(128) void pair_kernel(
    const int* __restrict__ atomic_numbers,
    const int* __restrict__ pair_indices,
    const float* __restrict__ r_ij,
    const float* __restrict__ d_ij,
    const float* __restrict__ emb_table,
    const float* __restrict__ w_zij, const float* __restrict__ b_zij,
    const float* __restrict__ w_I, const float* __restrict__ b_I,
    const float* __restrict__ w_A, const float* __restrict__ b_A,
    const float* __restrict__ w_S, const float* __restrict__ b_S,
    float* __restrict__ acc, float* __restrict__ outR)
{
  __shared__ float s_zfeat[4][32][64];  // concat(emb_i, emb_j) per pair (A of zij GEMM)
  __shared__ float s_radl[4][32][32];   // radial*rcut per pair (A of fI/fA/fS GEMMs)
  __shared__ float s_scal[4][32][10];   // x,y,z, sym6, rcut
  __shared__ int   s_atom[4][32];

  const int lane = threadIdx.x & 31;
  const int wv   = threadIdx.x >> 5;
  const int p    = blockIdx.x * 128 + wv * 32 + lane;   // 400000 = 3125*128, no tail

  // ---------------- per-pair scalar phase (1 pair per lane) ----------------
  const int ai = pair_indices[p];
  const int aj = pair_indices[N_PAIRS + p];
  s_atom[wv][lane] = ai;
  const int zi = atomic_numbers[ai];
  const int zj = atomic_numbers[aj];
  const float4* ei = (const float4*)(emb_table + (size_t)zi * 32);
  const float4* ej = (const float4*)(emb_table + (size_t)zj * 32);
  float4* zf = (float4*)(&s_zfeat[wv][lane][0]);
  #pragma unroll
  for (int q = 0; q < 8; ++q) zf[q] = ei[q];
  #pragma unroll
  for (int q = 0; q < 8; ++q) zf[8 + q] = ej[q];

  const float d    = d_ij[p];
  const float invd = 1.0f / d;
  const float x = r_ij[p * 3 + 0] * invd;
  const float y = r_ij[p * 3 + 1] * invd;
  const float z = r_ij[p * 3 + 2] * invd;
  const float rcut = (d < 0.5f) ? (0.5f * (cosf(d * 6.2831853071795865f) + 1.0f)) : 0.0f;
  const float t3 = (x * x + y * y + z * z) * (1.0f / 3.0f);
  s_scal[wv][lane][0] = x;
  s_scal[wv][lane][1] = y;
  s_scal[wv][lane][2] = z;
  s_scal[wv][lane][3] = x * x - t3;
  s_scal[wv][lane][4] = x * y;
  s_scal[wv][lane][5] = x * z;
  s_scal[wv][lane][6] = y * y - t3;
  s_scal[wv][lane][7] = y * z;
  s_scal[wv][lane][8] = z * z - t3;
  s_scal[wv][lane][9] = rcut;

  // radial basis * rcut  (alpha=0.1, R0=0, RC=0.5, N_RBF=32)
  const float e0    = expf(-d / 0.1f);
  const float start = expf(-5.0f);
  const float step  = (1.0f - start) / 31.0f;
  const float bb    = (2.0f / 32.0f) * (1.0f - start);
  const float beta  = 1.0f / (bb * bb);
  #pragma unroll
  for (int k = 0; k < 32; ++k) {
    const float c  = start + step * (float)k;
    const float df = e0 - c;
    const float rk = expf(-beta * df * df) * rcut;
    s_radl[wv][lane][k] = rk;
    outR[(size_t)p * 32 + k] = rk;   // second tuple output: radial[:,None,:]
  }
  __syncthreads();

  // ---------------- WMMA phase: f32 16x16x4 tiles ----------------
  const int half = lane >> 4;      // K half (A/B) and M half (C/D)
  const int nlo  = lane & 15;      // M row for A loads, N col for B/C/D
  const int koff = half * 2;

  #pragma unroll
  for (int mt = 0; mt < 2; ++mt) {
    const int arow = mt * 16 + nlo;
    #pragma unroll
    for (int nt = 0; nt < 2; ++nt) {
      const int n = nt * 16 + nlo;

      const float bz = b_zij[n];
      const float bi = b_I[n];
      const float ba = b_A[n];
      const float bs = b_S[n];
      v8f c8, fi8, fa8, fs8;
      #pragma unroll
      for (int v = 0; v < 8; ++v) { c8[v] = bz; fi8[v] = bi; fa8[v] = ba; fs8[v] = bs; }

      // zij = concat(emb) @ w_zij^T + b   (K = 64)
      #pragma unroll
      for (int k = 0; k < 16; ++k) {
        v2f a, b;
        a[0] = s_zfeat[wv][arow][4 * k + koff];
        a[1] = s_zfeat[wv][arow][4 * k + koff + 1];
        b[0] = w_zij[n * 64 + 4 * k + koff];
        b[1] = w_zij[n * 64 + 4 * k + koff + 1];
        c8 = wmma4(a, b, c8);
      }
      // fI/fA/fS = radial @ w^T + b   (K = 32)
      #pragma unroll
      for (int k = 0; k < 8; ++k) {
        v2f a, bI2, bA2, bS2;
        a[0] = s_radl[wv][arow][4 * k + koff];
        a[1] = s_radl[wv][arow][4 * k + koff + 1];
        bI2[0] = w_I[n * 32 + 4 * k + koff]; bI2[1] = w_I[n * 32 + 4 * k + koff + 1];
        bA2[0] = w_A[n * 32 + 4 * k + koff]; bA2[1] = w_A[n * 32 + 4 * k + koff + 1];
        bS2[0] = w_S[n * 32 + 4 * k + koff]; bS2[1] = w_S[n * 32 + 4 * k + koff + 1];
        fi8 = wmma4(a, bI2, fi8);
        fa8 = wmma4(a, bA2, fa8);
        fs8 = wmma4(a, bS2, fs8);
      }

      // scatter: D layout — lane half selects M+8, vgpr v is M row, n is column
      #pragma unroll
      for (int v = 0; v < 8; ++v) {
        const int m = mt * 16 + v + half * 8;    // pair row within this wave's 32 pairs
        const float xc  = s_scal[wv][m][0];
        const float yc  = s_scal[wv][m][1];
        const float zc  = s_scal[wv][m][2];
        const float sxx = s_scal[wv][m][3];
        const float sxy = s_scal[wv][m][4];
        const float sxz = s_scal[wv][m][5];
        const float syy = s_scal[wv][m][6];
        const float syz = s_scal[wv][m][7];
        const float szz = s_scal[wv][m][8];
        const float rc  = s_scal[wv][m][9];
        const int atom  = s_atom[wv][m];
        const float common = c8[v] * rc;      // C = rcut * zij
        const float vi = fi8[v] * common;
        const float va = fa8[v] * common;
        const float vs = fs8[v] * common;
        float* bp = acc + (size_t)atom * 320 + n;
        atomAddF(bp +   0, vi);
        atomAddF(bp +  32, va * xc);
        atomAddF(bp +  64, va * yc);
        atomAddF(bp +  96, va * zc);
        atomAddF(bp + 128, vs * sxx);
        atomAddF(bp + 160, vs * sxy);
        atomAddF(bp + 192, vs * sxz);
        atomAddF(bp + 224, vs * syy);
        atomAddF(bp + 256, vs * syz);
        atomAddF(bp + 288, vs * szz);
      }
    }
  }
}

__global__ __launch_bounds__(256) void atom_kernel(
    const float* __restrict__ acc,
    const float* __restrict__ w_t0, const float* __restrict__ w_t1, const float* __restrict__ w_t2,
    const float* __restrict__ w_s1, const float* __restrict__ b_s1,
    const float* __restrict__ w_s2, const float* __restrict__ b_s2,
    const float* __restrict__ ln_w, const float* __restrict__ ln_b,
    float* __restrict__ outX)
{
  __shared__ float s_wt0[1024], s_wt1[1024], s_wt2[1024];
  __shared__ float s_ws1[2048], s_ws2[6144];
  __shared__ float s_bs1[64], s_bs2[96], s_lnw[32], s_lnb[32];

  for (int i = threadIdx.x; i < 1024; i += 256) { s_wt0[i] = w_t0[i]; s_wt1[i] = w_t1[i]; s_wt2[i] = w_t2[i]; }
  for (int i = threadIdx.x; i < 2048; i += 256) s_ws1[i] = w_s1[i];
  for (int i = threadIdx.x; i < 6144; i += 256) s_ws2[i] = w_s2[i];
  if (threadIdx.x < 64) s_bs1[threadIdx.x] = b_s1[threadIdx.x];
  if (threadIdx.x < 96) s_bs2[threadIdx.x] = b_s2[threadIdx.x];
  if (threadIdx.x < 32) { s_lnw[threadIdx.x] = ln_w[threadIdx.x]; s_lnb[threadIdx.x] = ln_b[threadIdx.x]; }
  __syncthreads();

  const int a = blockIdx.x * 256 + threadIdx.x;
  if (a >= N_ATOMS) return;

  const float* R = acc + (size_t)a * 320;

  // norm2 per feature from structured accumulators: M = I*eye + skew(av) + sym(sv)
  float h[32];
  #pragma unroll
  for (int f = 0; f < 32; ++f) {
    const float i0 = R[f];
    const float ax = R[32 + f],  ay = R[64 + f],  az = R[96 + f];
    const float sxx = R[128 + f], sxy = R[160 + f], sxz = R[192 + f];
    const float syy = R[224 + f], syz = R[256 + f], szz = R[288 + f];
    const float m00 = i0 + sxx, m11 = i0 + syy, m22 = i0 + szz;
    const float m01 = sxy - az, m10 = sxy + az;
    const float m02 = sxz + ay, m20 = sxz - ay;
    const float m12 = syz - ax, m21 = syz + ax;
    h[f] = m00*m00 + m01*m01 + m02*m02 + m10*m10 + m11*m11 + m12*m12 + m20*m20 + m21*m21 + m22*m22;
  }
  // LayerNorm over F
  float mu = 0.0f;
  #pragma unroll
  for (int f = 0; f < 32; ++f) mu += h[f];
  mu *= (1.0f / 32.0f);
  float var = 0.0f;
  #pragma unroll
  for (int f = 0; f < 32; ++f) { const float t = h[f] - mu; var += t * t; }
  var *= (1.0f / 32.0f);
  const float rstd = 1.0f / sqrtf(var + 1e-5f);
  #pragma unroll
  for (int f = 0; f < 32; ++f) h[f] = (h[f] - mu) * rstd * s_lnw[f] + s_lnb[f];

  // h1 = silu(h @ w_s1^T + b_s1)
  float h1[64];
  #pragma unroll
  for (int g = 0; g < 64; ++g) {
    float s = s_bs1[g];
    #pragma unroll
    for (int f = 0; f < 32; ++f) s += h[f] * s_ws1[g * 32 + f];
    h1[g] = s / (1.0f + expf(-s));
  }

  // per output feature g: s3 gates + structured w_t transforms + assemble X
  for (int g = 0; g < 32; ++g) {
    float a0 = s_bs2[3 * g + 0], a1 = s_bs2[3 * g + 1], a2 = s_bs2[3 * g + 2];
    const float* w20 = s_ws2 + (size_t)(3 * g + 0) * 64;
    const float* w21 = s_ws2 + (size_t)(3 * g + 1) * 64;
    const float* w22 = s_ws2 + (size_t)(3 * g + 2) * 64;
    #pragma unroll
    for (int k = 0; k < 64; ++k) { const float hk = h1[k]; a0 += hk * w20[k]; a1 += hk * w21[k]; a2 += hk * w22[k]; }
    const float s0 = a0 / (1.0f + expf(-a0));
    const float s1 = a1 / (1.0f + expf(-a1));
    const float s2 = a2 / (1.0f + expf(-a2));

    const float* wr0 = s_wt0 + g * 32;
    const float* wr1 = s_wt1 + g * 32;
    const float* wr2 = s_wt2 + g * 32;
    float it = 0.0f, vax = 0.0f, vay = 0.0f, vaz = 0.0f;
    float t0 = 0.0f, t1 = 0.0f, t2 = 0.0f, t3v = 0.0f, t4 = 0.0f, t5 = 0.0f;
    #pragma unroll
    for (int f = 0; f < 32; ++f) {
      const float wi = wr0[f], wa = wr1[f], ws = wr2[f];
      it  += wi * R[f];
      vax += wa * R[32 + f];  vay += wa * R[64 + f];  vaz += wa * R[96 + f];
      t0  += ws * R[128 + f]; t1  += ws * R[160 + f]; t2  += ws * R[192 + f];
      t3v += ws * R[224 + f]; t4  += ws * R[256 + f]; t5  += ws * R[288 + f];
    }
    float* o = outX + ((size_t)a * 32 + g) * 9;
    o[0] = it * s0 + t0 * s2;
    o[1] = (-vaz) * s1 + t1 * s2;
    o[2] = ( vay) * s1 + t2 * s2;
    o[3] = ( vaz) * s1 + t1 * s2;
    o[4] = it * s0 + t3v * s2;
    o[5] = (-vax) * s1 + t4 * s2;
    o[6] = (-vay) * s1 + t2 * s2;
    o[7] = ( vax) * s1 + t4 * s2;
    o[8] = it * s0 + t5 * s2;
  }
}

extern "C" void kernel_launch(void* const* d_in, const int* in_sizes, int n_in,
                              void* d_out, int out_size, void* d_ws, size_t ws_size,
                              hipStream_t stream) {
  (void)in_sizes; (void)n_in; (void)out_size; (void)ws_size;
  const int*   atomic_numbers = (const int*)d_in[0];
  const int*   pair_indices   = (const int*)d_in[1];
  const float* r_ij   = (const float*)d_in[2];
  const float* d_ij   = (const float*)d_in[3];
  const float* emb    = (const float*)d_in[4];
  const float* w_zij  = (const float*)d_in[5];
  const float* b_zij  = (const float*)d_in[6];
  const float* w_I    = (const float*)d_in[7];
  const float* b_I    = (const float*)d_in[8];
  const float* w_A    = (const float*)d_in[9];
  const float* b_A    = (const float*)d_in[10];
  const float* w_S    = (const float*)d_in[11];
  const float* b_S    = (const float*)d_in[12];
  const float* w_t0   = (const float*)d_in[13];
  const float* w_t1   = (const float*)d_in[14];
  const float* w_t2   = (const float*)d_in[15];
  const float* w_s1   = (const float*)d_in[16];
  const float* b_s1   = (const float*)d_in[17];
  const float* w_s2   = (const float*)d_in[18];
  const float* b_s2   = (const float*)d_in[19];
  const float* ln_w   = (const float*)d_in[20];
  const float* ln_b   = (const float*)d_in[21];

  float* acc  = (float*)d_ws;                      // 32 MB structured accumulators
  float* outX = (float*)d_out;                     // (25000, 32, 3, 3)
  float* outR = outX + (size_t)N_ATOMS * 32 * 9;   // (400000, 1, 32)

  zero_ws_kernel<<<(ACC_FLOATS + 255) / 256, 256, 0, stream>>>(acc, ACC_FLOATS);
  pair_kernel<<<N_PAIRS / 128, 128, 0, stream>>>(
      atomic_numbers, pair_indices, r_ij, d_ij, emb,
      w_zij, b_zij, w_I, b_I, w_A, b_A, w_S, b_S, acc, outR);
  atom_kernel<<<(N_ATOMS + 255) / 256, 256, 0, stream>>>(
      acc, w_t0, w_t1, w_t2, w_s1, b_s1, w_s2, b_s2, ln_w, ln_b, outX);
}